// _System_robust_passive_input_54305566490948
// MI455X (gfx1250) — compile-verified
//
#include <hip/hip_runtime.h>
#include <hip/hip_bf16.h>

typedef float v2f __attribute__((ext_vector_type(2)));
typedef float v8f __attribute__((ext_vector_type(8)));

// D = A(16x4 f32) x B(4x16 f32) + C(16x16 f32), wave32 WMMA
#define WMMA4F32(a, b, c) \
  __builtin_amdgcn_wmma_f32_16x16x4_f32(false, (a), false, (b), (short)0, (c), false, false)

// Layout assumptions (CDNA5 ISA 7.12.2, 32-bit):
//  A frag (16x4):  lane L: M = L%16, elems i=0,1 hold K = 4*kc + 2*(L/16) + i
//  B frag (4x16):  lane L: N = L%16, elems i=0,1 hold K = 4*kc + 2*(L/16) + i
//  C/D (16x16):    lane L: N = L%16, VGPR r holds M = r + 8*(L/16)

__global__ __launch_bounds__(256) void ren_scan_kernel(
    const float* __restrict__ xi, const float* __restrict__ u,
    const float* __restrict__ C1, const float* __restrict__ D11,
    const float* __restrict__ D12, const float* __restrict__ bv,
    const float* __restrict__ A,  const float* __restrict__ B1,
    const float* __restrict__ B2, const float* __restrict__ bx,
    float* __restrict__ out)
{
  // per-block w/base staging tile: 128 rows x 128 q  (64 KB LDS)
  __shared__ float wS[128 * 128];

  const int lane = threadIdx.x & 31;
  const int wv   = threadIdx.x >> 5;          // 8 waves
  const int mlo  = lane & 15;                 // N (or M for A-frags)
  const int h    = lane >> 4;                 // lane-half
  const long rowBase = (long)blockIdx.x * 128 + (long)wv * 16; // 16 rows/wave
  float* wRow = &wS[(wv * 16) * 128];         // this wave's 16x128 region

  // ---------------- Stage A: base = xi@C1^T + u@D12^T + bv ----------------
  for (int t = 0; t < 8; ++t) {
    v8f acc = {};
    {
      const float* xa = xi + (rowBase + mlo) * 128 + 2 * h;
      const float* cb = C1 + (16 * t + mlo) * 128 + 2 * h;
      for (int kc = 0; kc < 32; ++kc) {
        v2f a = *(const v2f*)(xa + 4 * kc);
        v2f b = *(const v2f*)(cb + 4 * kc);
        acc = WMMA4F32(a, b, acc);
      }
    }
    {
      const float* ua = u   + (rowBase + mlo) * 64 + 2 * h;
      const float* db = D12 + (16 * t + mlo) * 64 + 2 * h;
      for (int kc = 0; kc < 16; ++kc) {
        v2f a = *(const v2f*)(ua + 4 * kc);
        v2f b = *(const v2f*)(db + 4 * kc);
        acc = WMMA4F32(a, b, acc);
      }
    }
    const float bvv = bv[16 * t + mlo];       // per-column bias (zeros here)
#pragma unroll
    for (int r = 0; r < 8; ++r)
      wRow[(r + 8 * h) * 128 + 16 * t + mlo] = acc[r] + bvv;
  }
  __syncthreads();

  // ------- Stage B: blocked forward substitution with tanh (w tile) -------
  for (int t = 0; t < 8; ++t) {
    // pending v for this 16-column block
    v8f acc;
#pragma unroll
    for (int r = 0; r < 8; ++r)
      acc[r] = wRow[(r + 8 * h) * 128 + 16 * t + mlo];

    // contributions from finished blocks s < t : w[:,s] @ D11[t,s]^T
    for (int s = 0; s < t; ++s) {
      const float* aw = wRow + mlo * 128 + 16 * s + 2 * h;          // LDS A-frag
      const float* bb = D11 + (16 * t + mlo) * 128 + 16 * s + 2 * h; // B-frag
#pragma unroll
      for (int kc = 0; kc < 4; ++kc) {
        v2f a = *(const v2f*)(aw + 4 * kc);
        v2f b = *(const v2f*)(bb + 4 * kc);
        acc = WMMA4F32(a, b, acc);
      }
    }
#pragma unroll
    for (int r = 0; r < 8; ++r)
      wRow[(r + 8 * h) * 128 + 16 * t + mlo] = acc[r];
    __syncthreads();

    // in-block sequential scan: lane m owns batch row m (no cross-lane dep)
    if (lane < 16) {
      float* wr = wRow + lane * 128 + 16 * t;
      for (int kk = 0; kk < 16; ++kk) {
        const float* drow = D11 + (16 * t + kk) * 128 + 16 * t;
        float v = wr[kk];                      // pending v (base + prior blocks)
        for (int j = 0; j < kk; ++j)
          v = fmaf(drow[j], wr[j], v);         // strictly-lower in-block part
        wr[kk] = tanhf(v);                     // finalize w
      }
    }
    __syncthreads();
  }

  // -------- Stage C: out = xi@A^T + w@B1^T + u@B2^T + bx ------------------
  for (int t = 0; t < 8; ++t) {
    v8f acc = {};
    {
      const float* xa = xi + (rowBase + mlo) * 128 + 2 * h;
      const float* ab = A  + (16 * t + mlo) * 128 + 2 * h;
      for (int kc = 0; kc < 32; ++kc) {
        v2f a = *(const v2f*)(xa + 4 * kc);
        v2f b = *(const v2f*)(ab + 4 * kc);
        acc = WMMA4F32(a, b, acc);
      }
    }
    {
      const float* wa = wRow + mlo * 128 + 2 * h;                   // LDS A-frag
      const float* b1 = B1 + (16 * t + mlo) * 128 + 2 * h;
      for (int kc = 0; kc < 32; ++kc) {
        v2f a = *(const v2f*)(wa + 4 * kc);
        v2f b = *(const v2f*)(b1 + 4 * kc);
        acc = WMMA4F32(a, b, acc);
      }
    }
    {
      const float* ua = u  + (rowBase + mlo) * 64 + 2 * h;
      const float* b2 = B2 + (16 * t + mlo) * 64 + 2 * h;
      for (int kc = 0; kc < 16; ++kc) {
        v2f a = *(const v2f*)(ua + 4 * kc);
        v2f b = *(const v2f*)(b2 + 4 * kc);
        acc = WMMA4F32(a, b, acc);
      }
    }
    const float bxv = bx[16 * t + mlo];
#pragma unroll
    for (int r = 0; r < 8; ++r)
      out[(rowBase + r + 8 * h) * 128 + 16 * t + mlo] = acc[r] + bxv;
  }
}

extern "C" void kernel_launch(void* const* d_in, const int* in_sizes, int n_in,
                              void* d_out, int out_size, void* d_ws, size_t ws_size,
                              hipStream_t stream) {
  const float* xi  = (const float*)d_in[0];
  const float* u   = (const float*)d_in[1];
  const float* C1  = (const float*)d_in[2];
  const float* D11 = (const float*)d_in[3];
  const float* D12 = (const float*)d_in[4];
  const float* bv  = (const float*)d_in[5];
  const float* A   = (const float*)d_in[6];
  const float* B1  = (const float*)d_in[7];
  const float* B2  = (const float*)d_in[8];
  const float* bx  = (const float*)d_in[9];
  float* out = (float*)d_out;

  const int Bn = in_sizes[0] / 128;          // batch rows
  const int grid = (Bn + 127) / 128;         // 128 rows per block (8 waves x 16)
  ren_scan_kernel<<<grid, 256, 0, stream>>>(xi, u, C1, D11, D12, bv, A, B1, B2, bx, out);
}